// GENEO_32538672235060
// MI455X (gfx1250) — compile-verified
//
#include <hip/hip_runtime.h>
#include <math.h>

typedef float v2f __attribute__((ext_vector_type(2)));
typedef float v8f __attribute__((ext_vector_type(8)));

// ---------------------------------------------------------------------------
// Pass 0: tiny prologue. off = floor(vectors) reduced mod H/W (so the hot
// gather pass needs no integer division), plus C and s constants.
// ---------------------------------------------------------------------------
__global__ void geneo_prep(const float* __restrict__ patterns,
                           const float* __restrict__ vectors,
                           int* __restrict__ offi, int* __restrict__ offj,
                           float* __restrict__ consts, int K, int pp,
                           int H, int W) {
  __shared__ float ps[64];
  const int t = threadIdx.x;
  if (t < K) {
    float s = 0.f;
    for (int e = 0; e < pp; ++e) s += patterns[(size_t)t * pp + e];
    ps[t] = s;
    int oi = (int)floorf(vectors[2 * t + 0]) % H; if (oi < 0) oi += H;
    int oj = (int)floorf(vectors[2 * t + 1]) % W; if (oj < 0) oj += W;
    offi[t] = oi;
    offj[t] = oj;
  }
  __syncthreads();
  if (t == 0) {
    float tot = 0.f;
    for (int k = 0; k < K; ++k) tot += ps[k];
    const float invpp = 1.0f / (float)pp;
    consts[0] = ((float)K - tot * invpp) / (float)(K - 1);  // C
    consts[1] = invpp / (float)(K - 1);                     // s
  }
}

// ---------------------------------------------------------------------------
// Pass 1: vertical band filter as WMMA:  colsum[i,j] = sum_{r=i-p..i-1} img[r,j]
// One wave per 16x16 tile; tile id readfirstlane'd into SGPRs.
// All element indices are 32-bit so loads/stores use SADDR + 32-bit voffset.
// ---------------------------------------------------------------------------
__global__ void __launch_bounds__(256)
geneo_vpass(const float* __restrict__ img, float* __restrict__ colsum,
            int W, int p, int tlog, int tilesX, int tiles) {
  const int wave = __builtin_amdgcn_readfirstlane(
      blockIdx.x * (blockDim.x >> 5) + (threadIdx.x >> 5));
  if (wave >= tiles) return;  // wave-uniform: EXEC all-ones for WMMA
  const int lane = threadIdx.x & 31;
  const int ln   = lane & 15;
  const int half = lane >> 4;
  const int ti = (tlog >= 0) ? (wave >> tlog) : (wave / tilesX);
  const int tj = wave - ti * tilesX;
  const int i0 = ti << 4, j0 = tj << 4;
  const unsigned col = (unsigned)(j0 + ln);
  const unsigned uW  = (unsigned)W;

  v8f acc = {0.f, 0.f, 0.f, 0.f, 0.f, 0.f, 0.f, 0.f};

  // A = U band (const), B = img rows i0-16..i0-1   (U[m,k] = k >= m+16-p)
#pragma unroll
  for (int c = 0; c < 4; ++c) {
    const int k = 4 * c + 2 * half;
    const int r = i0 - 16 + k;               // even, so r>=0 covers the pair
    const unsigned rc = (unsigned)((r < 0) ? 0 : r);  // clamped
    const float v0 = img[rc * uW + col];
    const float v1 = img[(rc + 1u) * uW + col];
    v2f a, b;
    a[0] = (k     >= ln + 16 - p) ? 1.f : 0.f;
    a[1] = (k + 1 >= ln + 16 - p) ? 1.f : 0.f;
    b[0] = (r >= 0) ? v0 : 0.f;
    b[1] = (r >= 0) ? v1 : 0.f;
    acc = __builtin_amdgcn_wmma_f32_16x16x4_f32(false, a, false, b, (short)0,
                                                acc, false, false);
  }
  // A = L band (const), B = img rows i0..i0+15     (L[m,k] = k<m && k>=m-p)
#pragma unroll
  for (int c = 0; c < 4; ++c) {
    const int k = 4 * c + 2 * half;
    const unsigned r = (unsigned)(i0 + k);
    v2f a, b;
    a[0] = ((k     < ln) && (k     >= ln - p)) ? 1.f : 0.f;
    a[1] = ((k + 1 < ln) && (k + 1 >= ln - p)) ? 1.f : 0.f;
    b[0] = img[r * uW + col];
    b[1] = img[(r + 1u) * uW + col];
    acc = __builtin_amdgcn_wmma_f32_16x16x4_f32(false, a, false, b, (short)0,
                                                acc, false, false);
  }
  const unsigned sbase = (unsigned)(i0 + 8 * half) * uW + col;
#pragma unroll
  for (int v = 0; v < 8; ++v)
    colsum[sbase + (unsigned)v * uW] = acc[v];
}

// ---------------------------------------------------------------------------
// Pass 2: horizontal band filter as WMMA:  win[i,j] = sum_{u=j-p..j-1} colsum[i,u]
// A-pairs are consecutive columns -> single 8B vector loads.
// ---------------------------------------------------------------------------
__global__ void __launch_bounds__(256)
geneo_hpass(const float* __restrict__ colsum, float* __restrict__ win,
            int W, int p, int tlog, int tilesX, int tiles) {
  const int wave = __builtin_amdgcn_readfirstlane(
      blockIdx.x * (blockDim.x >> 5) + (threadIdx.x >> 5));
  if (wave >= tiles) return;
  const int lane = threadIdx.x & 31;
  const int ln   = lane & 15;
  const int half = lane >> 4;
  const int ti = (tlog >= 0) ? (wave >> tlog) : (wave / tilesX);
  const int tj = wave - ti * tilesX;
  const int i0 = ti << 4, j0 = tj << 4;
  const unsigned uW  = (unsigned)W;
  const unsigned row = (unsigned)(i0 + ln) * uW;  // per-lane row base (32-bit)

  v8f acc = {0.f, 0.f, 0.f, 0.f, 0.f, 0.f, 0.f, 0.f};

  // A = colsum cols j0-16..j0-1, B = U band (const)
#pragma unroll
  for (int c = 0; c < 4; ++c) {
    const int k = 4 * c + 2 * half;
    const int cl = j0 - 16 + k;              // even; one predicate per pair
    const unsigned cc = (unsigned)((cl < 0) ? 0 : cl);  // clamped, 8B-aligned
    const v2f av = *(const v2f*)(colsum + (row + cc));
    v2f a, b;
    a[0] = (cl >= 0) ? av[0] : 0.f;
    a[1] = (cl >= 0) ? av[1] : 0.f;
    b[0] = (k     >= ln + 16 - p) ? 1.f : 0.f;
    b[1] = (k + 1 >= ln + 16 - p) ? 1.f : 0.f;
    acc = __builtin_amdgcn_wmma_f32_16x16x4_f32(false, a, false, b, (short)0,
                                                acc, false, false);
  }
  // A = colsum cols j0..j0+15, B = L band (const)
#pragma unroll
  for (int c = 0; c < 4; ++c) {
    const int k = 4 * c + 2 * half;
    const v2f a = *(const v2f*)(colsum + (row + (unsigned)(j0 + k)));
    v2f b;
    b[0] = ((k     < ln) && (k     >= ln - p)) ? 1.f : 0.f;
    b[1] = ((k + 1 < ln) && (k + 1 >= ln - p)) ? 1.f : 0.f;
    acc = __builtin_amdgcn_wmma_f32_16x16x4_f32(false, a, false, b, (short)0,
                                                acc, false, false);
  }
  const unsigned sbase = (unsigned)(i0 + 8 * half) * uW + (unsigned)(j0 + ln);
#pragma unroll
  for (int v = 0; v < 8; ++v)
    win[sbase + (unsigned)v * uW] = acc[v];
}

// ---------------------------------------------------------------------------
// Pass 3: K-way cyclic-shift gather-sum. win (16 MB) is L2-resident.
// Offsets are pre-reduced to [0,H)/[0,W): wrap = one compare + cond-add.
// ---------------------------------------------------------------------------
__global__ void __launch_bounds__(256)
geneo_gather(const float* __restrict__ win,
             const int* __restrict__ offi, const int* __restrict__ offj,
             const float* __restrict__ consts,
             float* __restrict__ out, int H, int W, int K) {
  __shared__ int soi[64], soj[64];
  __shared__ float sc[2];
  const int t = threadIdx.x;
  if (t < K) { soi[t] = offi[t]; soj[t] = offj[t]; }
  if (t < 2) sc[t] = consts[t];
  __syncthreads();

  const int j = blockIdx.x * blockDim.x + t;
  const int i = blockIdx.y;
  if (j >= W) return;
  const unsigned uW = (unsigned)W;

  float acc = 0.f;
#pragma unroll 4
  for (int k = 0; k < K; ++k) {
    int ri = i - soi[k]; if (ri < 0) ri += H;   // soi in [0,H)
    int rj = j - soj[k]; if (rj < 0) rj += W;   // soj in [0,W)
    acc += win[(unsigned)ri * uW + (unsigned)rj];
  }
  out[(unsigned)i * uW + (unsigned)j] = sc[0] + sc[1] * acc;
}

// ---------------------------------------------------------------------------
extern "C" void kernel_launch(void* const* d_in, const int* in_sizes, int n_in,
                              void* d_out, int out_size, void* d_ws, size_t ws_size,
                              hipStream_t stream) {
  const float* x        = (const float*)d_in[0];
  const float* patterns = (const float*)d_in[1];
  const float* vectors  = (const float*)d_in[2];
  float* out = (float*)d_out;

  const int K  = in_sizes[2] / 2;            // 16
  const int pp = in_sizes[1] / K;            // p*p = 225
  const int p  = (int)(sqrt((double)pp) + 0.5);
  const int H  = (int)(sqrt((double)in_sizes[0]) + 0.5);
  const int W  = H;

  // Buffers: colsum reuses d_out (consumed by pass 2 before pass 3 writes it);
  // win lives in d_ws, followed by the small header.
  float* colsum = out;
  float* win    = (float*)d_ws;
  char*  hdr    = (char*)d_ws + (size_t)H * W * sizeof(float);
  int*   offi   = (int*)hdr;
  int*   offj   = offi + 64;
  float* consts = (float*)(offj + 64);

  geneo_prep<<<1, 64, 0, stream>>>(patterns, vectors, offi, offj, consts,
                                   K, pp, H, W);

  const int tilesX = W / 16, tilesY = H / 16;
  const int tiles  = tilesX * tilesY;                 // 16384
  const int tlog   = ((tilesX & (tilesX - 1)) == 0) ? __builtin_ctz(tilesX) : -1;
  const int blocks = (tiles + 7) / 8;                 // 8 waves (tiles) / block
  geneo_vpass<<<blocks, 256, 0, stream>>>(x, colsum, W, p, tlog, tilesX, tiles);
  geneo_hpass<<<blocks, 256, 0, stream>>>(colsum, win, W, p, tlog, tilesX, tiles);

  dim3 g((W + 255) / 256, H);
  geneo_gather<<<g, 256, 0, stream>>>(win, offi, offj, consts, out, H, W, K);
}